// DinoPool_36404142801318
// MI455X (gfx1250) — compile-verified
//
#include <hip/hip_runtime.h>

// DinoPool: block-diagonal masked attention with uniform scores == per-block
// masked average pooling over V.  B=4, S=4096, H*D=384, BLOCK=512.
//
//   y[b, s, c] = (1/512) * sum_{k in block(s)} x[b, k, c]
//
// Phase 1: per (tile, 128-row chunk) exact f32 column sums via
//          V_WMMA_F32_16X16X4_F32 with an all-ones A matrix -> ws partials.
// Phase 2: combine 4 partials, scale, broadcast 64 rows per workgroup.
// Two kernels so the bandwidth-bound streaming read/write phases are spread
// over 128/256 workgroups instead of 32 (MI455X has far more WGPs than 32;
// 23.3 TB/s HBM / 192 MB L2 need wide parallelism to saturate).

typedef __attribute__((ext_vector_type(2))) float v2f;
typedef __attribute__((ext_vector_type(4))) float v4f;
typedef __attribute__((ext_vector_type(8))) float v8f;

#define DP_CH    384                   // H*D
#define DP_BLOCK 512                   // mask window
#define DP_SEQ   4096
#define DP_NBLK  (DP_SEQ / DP_BLOCK)   // 8 blocks per batch
#define DP_TILES (4 * DP_NBLK)         // 32 (batch, block) tiles
#define DP_CHUNK 128                   // rows reduced per phase-1 workgroup
#define DP_NCHNK (DP_BLOCK / DP_CHUNK) // 4 partials per tile
#define DP_WROWS 64                    // rows written per phase-2 workgroup
#define DP_C4    (DP_CH / 4)           // 96 float4 per row

// ---------------------------------------------------------------- phase 1 ---
__launch_bounds__(256)
__global__ void dino_pool_reduce_kernel(const float* __restrict__ x,
                                        float* __restrict__ partial) {
  const int tile  = blockIdx.x >> 2;       // 0..31
  const int chunk = blockIdx.x & 3;        // 0..3
  const int b     = tile / DP_NBLK;
  const int blk   = tile % DP_NBLK;
  const float* __restrict__ xt =
      x + ((size_t)b * DP_SEQ + (size_t)blk * DP_BLOCK + (size_t)chunk * DP_CHUNK) * DP_CH;

  const int tid  = threadIdx.x;
  const int wave = tid >> 5;               // 0..7
  const int lane = tid & 31;
  const int n    = lane & 15;              // B/D column for this lane
  const int kh   = (lane >> 4) << 1;       // row-half: 0 (lanes 0-15) / 2 (16-31)

  // Each wave owns 3 groups of 16 channels (8 waves * 3 * 16 = 384).
  const int g0  = wave * 3;
  const int cg0 = (g0 + 0) * 16 + n;
  const int cg1 = (g0 + 1) * 16 + n;
  const int cg2 = (g0 + 2) * 16 + n;

  // A = all-ones 16x4 f32 -> D[m,n] = sum_k B[k,n] (exact f32 column sums).
  v2f a; a.x = 1.0f; a.y = 1.0f;
  v8f c0 = {}; v8f c1 = {}; v8f c2 = {};

  // 128 rows, 4 per WMMA, 3 independent accumulators for ILP.
  // B layout (4x16 f32): VGPR0 = row kh, VGPR1 = row kh+1.
  for (int r = 0; r < DP_CHUNK; r += 4) {
    const float* p0 = xt + (size_t)(r + kh) * DP_CH;
    const float* p1 = p0 + DP_CH;
    __builtin_prefetch(p0 + 4 * DP_CH, 0, 3);   // global_prefetch_b8
    v2f b0, b1, b2;
    b0.x = p0[cg0]; b0.y = p1[cg0];
    b1.x = p0[cg1]; b1.y = p1[cg1];
    b2.x = p0[cg2]; b2.y = p1[cg2];
    c0 = __builtin_amdgcn_wmma_f32_16x16x4_f32(false, a, false, b0,
                                               (short)0, c0, false, false);
    c1 = __builtin_amdgcn_wmma_f32_16x16x4_f32(false, a, false, b1,
                                               (short)0, c1, false, false);
    c2 = __builtin_amdgcn_wmma_f32_16x16x4_f32(false, a, false, b2,
                                               (short)0, c2, false, false);
  }

  // All D rows are equal column sums; lane (L&15) holds column L&15.
  float* ps = partial + ((size_t)tile * DP_NCHNK + chunk) * DP_CH;
  if (lane < 16) {
    ps[cg0] = c0[0];
    ps[cg1] = c1[0];
    ps[cg2] = c2[0];
  }
}

// ---------------------------------------------------------------- phase 2 ---
__launch_bounds__(256)
__global__ void dino_pool_bcast_kernel(const float* __restrict__ partial,
                                       float* __restrict__ y) {
  const int tile = blockIdx.x >> 3;        // 0..31
  const int rc   = blockIdx.x & 7;         // 0..7 (64-row slab within block)
  const int b    = tile / DP_NBLK;
  const int blk  = tile % DP_NBLK;

  __shared__ __align__(16) float mean_lds[DP_CH];

  const float* ps = partial + (size_t)tile * DP_NCHNK * DP_CH;
  const int tid = threadIdx.x;
  const float inv = 1.0f / (float)DP_BLOCK;
  for (int c = tid; c < DP_CH; c += 256) {
    float s = ps[c] + ps[DP_CH + c] + ps[2 * DP_CH + c] + ps[3 * DP_CH + c];
    mean_lds[c] = s * inv;
  }
  __syncthreads();

  float* yt = y + ((size_t)b * DP_SEQ + (size_t)blk * DP_BLOCK +
                   (size_t)rc * DP_WROWS) * DP_CH;
  const v4f* __restrict__ m4 = (const v4f*)mean_lds;
  v4f*       __restrict__ y4 = (v4f*)yt;
  // 64 rows * 96 float4 = 6144 vec4 stores, coalesced.
  for (int idx = tid; idx < DP_WROWS * DP_C4; idx += 256) {
    y4[idx] = m4[idx % DP_C4];
  }
}

// ------------------------------------------------------------------ launch ---
extern "C" void kernel_launch(void* const* d_in, const int* in_sizes, int n_in,
                              void* d_out, int out_size, void* d_ws, size_t ws_size,
                              hipStream_t stream) {
  (void)in_sizes; (void)n_in; (void)out_size; (void)ws_size;
  const float* x = (const float*)d_in[0];   // [4, 4096, 384] f32
  // d_in[1], d_in[2] (x1, x2) unused by the module; d_in[3] mask structure is
  // the fixed 512-wide block diagonal, baked into the kernel.
  float* y       = (float*)d_out;           // [4, 4096, 384] f32
  float* partial = (float*)d_ws;            // [32][4][384] f32 = 192 KB

  dino_pool_reduce_kernel<<<DP_TILES * DP_NCHNK, 256, 0, stream>>>(x, partial);
  dino_pool_bcast_kernel <<<DP_TILES * 8,        256, 0, stream>>>(partial, y);
}